// PcheckArcFaceInnerProduct_20100446945296
// MI455X (gfx1250) — compile-verified
//
#include <hip/hip_runtime.h>
#include <hip/hip_bf16.h>
#include <math.h>

typedef __attribute__((ext_vector_type(2))) float v2f;
typedef __attribute__((ext_vector_type(8))) float v8f;

#define B_ROWS 256
#define D_DIM  128
#define C_COLS 100000
#define N_TILES (C_COLS / 16)       // 6250
#define WAVES_PER_BLOCK 4

#define SCALE_S  35.0f
#define COS_M    0.8775825618903728f   // cos(0.5)
#define SIN_M    0.479425538604203f    // sin(0.5)
#define THRESH  (-0.8775825618903728f) // -cos(0.5)
#define EXT_VAL (-0.2397127693021015f) // -0.5*sin(0.5)

// ---------------------------------------------------------------------------
// Kernel 1: reciprocal L2 norm of each 128-float row. One wave32 per row:
// each lane loads a float4 (32 lanes * 4 = 128 floats), shfl-xor reduce.
// ---------------------------------------------------------------------------
__global__ void row_rnorm_kernel(const float* __restrict__ x,
                                 float* __restrict__ out, int nrows) {
    int gtid = blockIdx.x * blockDim.x + threadIdx.x;
    int row  = gtid >> 5;
    int lane = threadIdx.x & 31;
    if (row >= nrows) return;

    const float4* p = (const float4*)(x + (size_t)row * D_DIM);
    float4 v = p[lane];
    float s = v.x * v.x + v.y * v.y + v.z * v.z + v.w * v.w;
#pragma unroll
    for (int off = 16; off > 0; off >>= 1)
        s += __shfl_xor(s, off, 32);
    if (lane == 0)
        out[row] = 1.0f / sqrtf(s);
}

// ---------------------------------------------------------------------------
// Kernel 2: fused GEMM (inner = feat @ weights^T via V_WMMA_F32_16X16X4_F32)
// + cosine normalization + ArcFace margin epilogue + 3 output stores.
// One wave per 16x16 output tile; K=128 -> 32 WMMA ops.
// ---------------------------------------------------------------------------
__global__ void __launch_bounds__(WAVES_PER_BLOCK * 32)
arcface_fused_kernel(const float* __restrict__ feat,
                     const float* __restrict__ weights,
                     const int*   __restrict__ label,
                     const float* __restrict__ rnw,   // [C] 1/||w_c||
                     const float* __restrict__ rnf,   // [B] 1/||f_b||
                     float* __restrict__ out) {
    const int lane = threadIdx.x & 31;
    const int wave = threadIdx.x >> 5;
    const int ntile = blockIdx.x * WAVES_PER_BLOCK + wave;
    if (ntile >= N_TILES) return;          // wave-uniform: EXEC stays all-1s

    const int c0 = ntile * 16;
    const int m0 = blockIdx.y * 16;
    const int l15  = lane & 15;
    const int half = lane >> 4;            // 0: K pair {k,k+1}, 1: {k+2,k+3}

    // A lane pointer: feat[m0 + l15][2*half + k], B lane pointer: weights row c0+l15
    const float* arow = feat    + (size_t)(m0 + l15) * D_DIM + 2 * half;
    const float* brow = weights + (size_t)(c0 + l15) * D_DIM + 2 * half;

    v8f acc = {};
#pragma unroll
    for (int k = 0; k < D_DIM; k += 4) {
        v2f a = *(const v2f*)(arow + k);   // 8-byte aligned
        v2f b = *(const v2f*)(brow + k);
        acc = __builtin_amdgcn_wmma_f32_16x16x4_f32(
            false, a, false, b, (short)0, acc, false, false);
    }

    // ---- epilogue ----
    const int   cg_col = c0 + l15;         // this lane's class column
    const float rnw_c  = rnw[cg_col];

    const size_t BC = (size_t)B_ROWS * C_COLS;
    float* __restrict__ out_cos   = out;
    float* __restrict__ out_marg  = out + BC;
    float* __restrict__ out_inner = out + 2 * BC;

#pragma unroll
    for (int r = 0; r < 8; ++r) {
        const int   m     = m0 + r + 8 * half;   // row for this VGPR slot
        const float inner = acc[r];
        const float cosv  = inner * rnf[m] * rnw_c;

        float marg = SCALE_S * cosv;
        if (label[m] == cg_col) {
            float cg = fminf(fmaxf(cosv, -1.0f), 1.0f);
            float sg = sqrtf(fmaxf(1.0f - cg * cg, 0.0f));
            float gm = (cg > THRESH) ? (cg * COS_M - sg * SIN_M)
                                     : (cg + EXT_VAL);
            marg = SCALE_S * gm;
        }

        const size_t idx = (size_t)m * C_COLS + cg_col;
        out_cos[idx]   = cosv;
        out_marg[idx]  = marg;
        out_inner[idx] = inner;
    }
}

// ---------------------------------------------------------------------------
extern "C" void kernel_launch(void* const* d_in, const int* in_sizes, int n_in,
                              void* d_out, int out_size, void* d_ws, size_t ws_size,
                              hipStream_t stream) {
    const float* feat    = (const float*)d_in[0];   // [256,128]
    const float* weights = (const float*)d_in[1];   // [100000,128]
    const int*   label   = (const int*)d_in[2];     // [256]
    float*       out     = (float*)d_out;           // 3 * 256 * 100000 f32

    // workspace: [0, C) = 1/||w_c||, [C, C+B) = 1/||f_b||
    float* rnw = (float*)d_ws;
    float* rnf = rnw + C_COLS;

    // 8 waves per 256-thread block -> 8 rows per block
    {
        int rows = C_COLS;
        int blocks = (rows * 32 + 255) / 256;
        row_rnorm_kernel<<<blocks, 256, 0, stream>>>(weights, rnw, rows);
    }
    {
        int rows = B_ROWS;
        int blocks = (rows * 32 + 255) / 256;
        row_rnorm_kernel<<<blocks, 256, 0, stream>>>(feat, rnf, rows);
    }

    dim3 grid((N_TILES + WAVES_PER_BLOCK - 1) / WAVES_PER_BLOCK, B_ROWS / 16);
    dim3 block(WAVES_PER_BLOCK * 32);
    arcface_fused_kernel<<<grid, block, 0, stream>>>(feat, weights, label,
                                                     rnw, rnf, out);
}